// AnomalyMapGenerator_23991687315897
// MI455X (gfx1250) — compile-verified
//
#include <hip/hip_runtime.h>
#include <math.h>

// ---------------------------------------------------------------------------
// AnomalyMapGenerator for MI455X (gfx1250, wave32)
//
//  distance [16,3136,4096] f32 (822 MB)  --HBM-bound scan-->  score [16,56,56]
//  score --(M . S . M^T, M = Blur(33,sigma4,reflect) o BilinearResize)--> out [16,1,224,224]
//
//  Stage 1: one wave per row, b128 non-temporal streaming loads, branchless
//           running top-3, shfl_xor butterfly merge, softmin score.
//  Stage 2: precompute M (224x56); two fp32 WMMA GEMMs per image using
//           V_WMMA_F32_16X16X4_F32. Score stored transposed + zero-padded so
//           every fragment load is an unconditional aligned b64 (no EXEC
//           divergence in the GEMM loops).
// ---------------------------------------------------------------------------

typedef __attribute__((ext_vector_type(2))) float v2f;
typedef __attribute__((ext_vector_type(4))) float v4f;
typedef __attribute__((ext_vector_type(8))) float v8f;

#define NB      16
#define HWROWS  3136
#define MCOLS   4096
#define H_IN    56
#define NPAD    64      // padded N for 4x 16-wide tiles
#define H_OUT   224
#define KSIZE   33
#define KPAD    16
#define SIGMA2X 32.0f   // 2*sigma^2, sigma=4

// ws layout (floats)
#define OFF_M   0                       // Mmat  : 224*56          = 12544
#define OFF_ST  12544                   // ST    : 16*64*56        = 57344
#define OFF_T   (12544 + 57344)         // Tpad  : 16*224*64       = 229376

// ---------------------------------------------------------------------------
// Stage 2a: build the fused operator M[224][56] = G(blur,reflect) @ R(bilinear)
// ---------------------------------------------------------------------------
__global__ void build_M_kernel(float* __restrict__ Mmat) {
    int idx = blockIdx.x * blockDim.x + threadIdx.x;   // exactly 224*56 threads
    int i = idx / H_IN;      // output row   (0..223)
    int j = idx % H_IN;      // input column (0..55)

    float gsum = 0.0f;
    #pragma unroll
    for (int t = 0; t < KSIZE; ++t) {
        float x = (float)(t - KPAD);
        gsum += __expf(-(x * x) / SIGMA2X);
    }

    float acc = 0.0f;
    #pragma unroll
    for (int t = 0; t < KSIZE; ++t) {
        float x = (float)(t - KPAD);
        float g = __expf(-(x * x) / SIGMA2X);
        int k = i + t - KPAD;                 // reflect padding (no edge repeat)
        if (k < 0)    k = -k;
        if (k > 223)  k = 446 - k;
        // bilinear 56->224, half-pixel centers: src = (k+0.5)/4 - 0.5
        float src = (k + 0.5f) * 0.25f - 0.5f;
        float ff  = floorf(src);
        float w1  = src - ff;
        int f0 = (int)ff;
        int j0 = min(max(f0, 0), H_IN - 1);
        int j1 = min(max(f0 + 1, 0), H_IN - 1);
        float r = 0.0f;
        if (j0 == j) r += 1.0f - w1;
        if (j1 == j) r += w1;
        acc += g * r;
    }
    Mmat[idx] = acc / gsum;
}

// zero the padded, transposed score buffer (pad columns must be exact zeros)
__global__ void zero_ST_kernel(float* __restrict__ ST) {
    ST[blockIdx.x * blockDim.x + threadIdx.x] = 0.0f;   // exactly 16*64*56
}

// ---------------------------------------------------------------------------
// Stage 1: top-3 + softmin score. One wave32 per row of 4096 floats.
// ---------------------------------------------------------------------------
__device__ __forceinline__ void insert3(float& m0, float& m1, float& m2, float x) {
    float a = fminf(m0, x);
    float b = fmaxf(m0, x);
    m0 = a;
    float c = fminf(m1, b);
    float d = fmaxf(m1, b);
    m1 = c;
    m2 = fminf(m2, d);
}

__device__ __forceinline__ void merge3(float& a0, float& a1, float& a2,
                                       float b0, float b1, float b2) {
    float c0, c1, c2;
    if (a0 <= b0) {
        c0 = a0;
        if (a1 <= b0) { c1 = a1; c2 = fminf(a2, b0); }
        else          { c1 = b0; c2 = fminf(a1, b1); }
    } else {
        c0 = b0;
        if (b1 <= a0) { c1 = b1; c2 = fminf(b2, a0); }
        else          { c1 = a0; c2 = fminf(a1, b1); }
    }
    a0 = c0; a1 = c1; a2 = c2;
}

__global__ void __launch_bounds__(256)
topk_score_kernel(const float* __restrict__ dist, float* __restrict__ ST) {
    const int lane = threadIdx.x & 31;
    const int wave = threadIdx.x >> 5;
    const int row  = blockIdx.x * 8 + wave;           // grid sized exactly

    const v4f* p = reinterpret_cast<const v4f*>(dist + (size_t)row * MCOLS) + lane;

    float m0 = 3.4e38f, m1 = 3.4e38f, m2 = 3.4e38f;
    #pragma unroll 4
    for (int c = 0; c < 32; ++c) {
        v4f v = __builtin_nontemporal_load(p + c * 32);   // b128 NT stream
        if (c + 8 < 32)
            __builtin_prefetch((const void*)(p + (c + 8) * 32), 0, 0);
        float lo = fminf(fminf(v.x, v.y), fminf(v.z, v.w));
        if (lo < m2) {                                   // wave-level skip (execz)
            insert3(m0, m1, m2, v.x);
            insert3(m0, m1, m2, v.y);
            insert3(m0, m1, m2, v.z);
            insert3(m0, m1, m2, v.w);
        }
    }

    #pragma unroll
    for (int mask = 16; mask >= 1; mask >>= 1) {
        float b0 = __shfl_xor(m0, mask, 32);
        float b1 = __shfl_xor(m1, mask, 32);
        float b2 = __shfl_xor(m2, mask, 32);
        merge3(m0, m1, m2, b0, b1, b2);
    }

    if (lane == 0) {
        float v0 = sqrtf(m0), v1 = sqrtf(m1), v2 = sqrtf(m2);
        float w0 = 1.0f / (1.0f + __expf(v0 - v1) + __expf(v0 - v2));
        // transposed store: ST[b][w][h] = s(b, h, w); b = row/3136, hw = row%3136
        int b  = row / HWROWS;
        int hw = row - b * HWROWS;
        int h  = hw / H_IN;
        int w  = hw - h * H_IN;
        ST[b * (NPAD * H_IN) + w * H_IN + h] = w0 * v0;
    }
}

// ---------------------------------------------------------------------------
// V_WMMA_F32_16X16X4_F32 fragments (wave32):
//   A 16x4 : lanes 0-15 -> M=lane, K={0,1}; lanes 16-31 -> K={2,3}
//   B 4x16 : VGPR0: K={0|2} row striped over lanes; VGPR1: K={1|3}
//   C/D    : VGPR r: lanes 0-15 -> M=r, N=lane; lanes 16-31 -> M=r+8
// All loads below are contiguous aligned v2f (b64), no predication.
// ---------------------------------------------------------------------------

// Stage 2b: Tpad[b] (224x64) = M (224x56) . S[b] (56x64, cols 56..63 zero)
__global__ void __launch_bounds__(128)
gemm_MS_kernel(const float* __restrict__ Mmat, const float* __restrict__ ST,
               float* __restrict__ Tpad) {
    const int lane = threadIdx.x & 31;
    const int tile = blockIdx.x * 4 + (threadIdx.x >> 5);   // 896 tiles exactly
    const int b  = tile / 56;
    const int t2 = tile % 56;
    const int row0 = (t2 / 4) * 16;      // 14 M-tiles
    const int col0 = (t2 % 4) * 16;      // 4 N-tiles over padded 64
    const int l  = lane & 15;
    const int hi = lane >> 4;
    const int col = col0 + l;

    const float* Arow = Mmat + (row0 + l) * H_IN + 2 * hi;
    const float* Bcol = ST + b * (NPAD * H_IN) + col * H_IN + 2 * hi;  // S^T[col][*]

    v8f acc = {};
    #pragma unroll
    for (int k = 0; k < H_IN; k += 4) {
        v2f a  = *reinterpret_cast<const v2f*>(Arow + k);
        v2f bb = *reinterpret_cast<const v2f*>(Bcol + k);
        acc = __builtin_amdgcn_wmma_f32_16x16x4_f32(
            false, a, false, bb, (short)0, acc, false, false);
    }

    float* Tb = Tpad + b * (H_OUT * NPAD);
    #pragma unroll
    for (int r = 0; r < 8; ++r)
        Tb[(row0 + r + 8 * hi) * NPAD + col] = acc[r];   // pad cols hold exact 0
}

// Stage 2c: out[b] (224x224) = Tpad[b] (224x56 valid) . M^T
__global__ void __launch_bounds__(128)
gemm_TMt_kernel(const float* __restrict__ Tpad, const float* __restrict__ Mmat,
                float* __restrict__ out) {
    const int lane = threadIdx.x & 31;
    const int tile = blockIdx.x * 4 + (threadIdx.x >> 5);   // 3136 tiles exactly
    const int b  = tile / 196;
    const int t2 = tile % 196;
    const int row0 = (t2 / 14) * 16;
    const int col0 = (t2 % 14) * 16;
    const int l  = lane & 15;
    const int hi = lane >> 4;

    const float* Arow = Tpad + b * (H_OUT * NPAD) + (row0 + l) * NPAD + 2 * hi;
    const float* Brow = Mmat + (col0 + l) * H_IN + 2 * hi;   // B = M^T

    v8f acc = {};
    #pragma unroll
    for (int k = 0; k < H_IN; k += 4) {
        v2f a  = *reinterpret_cast<const v2f*>(Arow + k);
        v2f bb = *reinterpret_cast<const v2f*>(Brow + k);
        acc = __builtin_amdgcn_wmma_f32_16x16x4_f32(
            false, a, false, bb, (short)0, acc, false, false);
    }

    float* ob = out + (size_t)b * H_OUT * H_OUT;
    #pragma unroll
    for (int r = 0; r < 8; ++r)
        ob[(row0 + r + 8 * hi) * H_OUT + col0 + l] = acc[r];
}

// ---------------------------------------------------------------------------
extern "C" void kernel_launch(void* const* d_in, const int* in_sizes, int n_in,
                              void* d_out, int out_size, void* d_ws, size_t ws_size,
                              hipStream_t stream) {
    const float* dist = (const float*)d_in[0];   // [16,3136,4096] f32
    float* out = (float*)d_out;                  // [16,1,224,224] f32
    float* ws  = (float*)d_ws;

    float* Mmat = ws + OFF_M;
    float* ST   = ws + OFF_ST;
    float* Tpad = ws + OFF_T;

    build_M_kernel   <<<49,   256, 0, stream>>>(Mmat);          // 12544 threads
    zero_ST_kernel   <<<224,  256, 0, stream>>>(ST);            // 57344 threads
    topk_score_kernel<<<6272, 256, 0, stream>>>(dist, ST);      // 50176 waves
    gemm_MS_kernel   <<<224,  128, 0, stream>>>(Mmat, ST, Tpad);   // 896 tiles
    gemm_TMt_kernel  <<<784,  128, 0, stream>>>(Tpad, Mmat, out);  // 3136 tiles
}